// GAT_45483703665320
// MI455X (gfx1250) — compile-verified
//
#include <hip/hip_runtime.h>
#include <math.h>

typedef float v2f __attribute__((ext_vector_type(2)));
typedef float v8f __attribute__((ext_vector_type(8)));

#define NEG_SLOPE 0.2f
#define BPAD 260            // padded dword stride per column (256 + 4)
#define KMAX 256            // K is 256 for all three layers

// ---------------------------------------------------------------------------
// WMMA fp32 GEMM: C[M,Nc] = A[M,K] @ B[K,Nc], row-major, K % 4 == 0, K <= 256.
// 256 threads = 8 waves stacked in M; block tile = 128 rows x 64 cols.
// B tile staged in LDS column-major (Bs[col*BPAD + k], pad 4 -> the 64 dword
// accesses of each ds_load_b64 hit 64 distinct banks). Each wave computes a
// 16x64 strip: 4 v_wmma_f32_16x16x4_f32 accumulators reuse one A fragment.
// Inner loop per K-step: 1 global b64 (A) + 4 ds_load_b64 (B) + 4 WMMA,
// all offsets immediate -> no address math in the loop.
// Out-of-range rows/cols are CLAMPED, not masked: garbage A row m / B col n
// only pollutes output row m / col n, which the guarded epilogue never stores.
// ---------------------------------------------------------------------------
__global__ __launch_bounds__(256)
void gat_gemm_f32_wmma(const float* __restrict__ A, const float* __restrict__ B,
                       float* __restrict__ C, int M, int K, int Nc) {
  __shared__ float Bs[64 * BPAD];

  const int tid  = threadIdx.x;
  const int lane = tid & 31;
  const int wave = tid >> 5;
  const int m0   = blockIdx.x * 128 + wave * 16;
  const int n0   = blockIdx.y * 64;
  const int half = lane >> 4;        // ISA A/B layout: lanes16-31 carry K+2,K+3
  const int l16  = lane & 15;

  // ---- stage B tile (K rows x 64 cols) into LDS, column-major ----
  {
    const int col   = tid & 63;
    const int colg  = n0 + col;
    const int col_c = colg < Nc ? colg : Nc - 1;   // clamp, see header note
    const float* __restrict__ Bg = B + col_c;
    float* __restrict__ Bd = Bs + col * BPAD;
    for (int k = (tid >> 6); k < K; k += 4)        // coalesced b32 per round
      Bd[k] = Bg[(size_t)k * Nc];
  }
  __syncthreads();

  const int ra   = m0 + l16;
  const int ra_c = ra < M ? ra : M - 1;
  const float* __restrict__ Ap = A + (size_t)ra_c * K + 2 * half;
  const float* __restrict__ Bl = Bs + l16 * BPAD + 2 * half;

  v8f acc0 = {}, acc1 = {}, acc2 = {}, acc3 = {};
  #pragma unroll 4
  for (int kk = 0; kk < K; kk += 4) {
    v2f a  = *reinterpret_cast<const v2f*>(Ap + kk);
    v2f b0 = *reinterpret_cast<const v2f*>(Bl + kk);
    v2f b1 = *reinterpret_cast<const v2f*>(Bl + 16 * BPAD + kk);
    v2f b2 = *reinterpret_cast<const v2f*>(Bl + 32 * BPAD + kk);
    v2f b3 = *reinterpret_cast<const v2f*>(Bl + 48 * BPAD + kk);
    acc0 = __builtin_amdgcn_wmma_f32_16x16x4_f32(false, a, false, b0,
                                                 (short)0, acc0, false, false);
    acc1 = __builtin_amdgcn_wmma_f32_16x16x4_f32(false, a, false, b1,
                                                 (short)0, acc1, false, false);
    acc2 = __builtin_amdgcn_wmma_f32_16x16x4_f32(false, a, false, b2,
                                                 (short)0, acc2, false, false);
    acc3 = __builtin_amdgcn_wmma_f32_16x16x4_f32(false, a, false, b3,
                                                 (short)0, acc3, false, false);
  }

  #pragma unroll
  for (int r = 0; r < 8; ++r) {        // D layout: VGPR r -> rows r, r+8
    const int row = m0 + r + 8 * half;
    if (row < M) {
      float* __restrict__ Cr = C + (size_t)row * Nc;
      const int cb = n0 + l16;
      if (cb      < Nc) Cr[cb]      = acc0[r];
      if (cb + 16 < Nc) Cr[cb + 16] = acc1[r];
      if (cb + 32 < Nc) Cr[cb + 32] = acc2[r];
      if (cb + 48 < Nc) Cr[cb + 48] = acc3[r];
    }
  }
}

// --------------------------- attention helpers -----------------------------
__device__ __forceinline__ float lrelu(float x) {
  return x > 0.f ? x : NEG_SLOPE * x;
}

// Float atomic max via signed/unsigned integer ordering (init must be -inf).
__device__ __forceinline__ void atomicMaxF(float* addr, float v) {
  if (v >= 0.f) atomicMax((int*)addr, __float_as_int(v));
  else          atomicMin((unsigned int*)addr, __float_as_uint(v));
}

// el/er per (node, head): dot(h[n,h,:], al[h,:]) / dot(h[n,h,:], ar[h,:])
__global__ void gat_attn_coeff(const float* __restrict__ h,
                               const float* __restrict__ al,
                               const float* __restrict__ ar,
                               float* __restrict__ el, float* __restrict__ er,
                               int Nn, int H, int D) {
  int t = blockIdx.x * blockDim.x + threadIdx.x;
  if (t >= Nn * H) return;
  int n = t / H, hh = t % H;
  const float* hp  = h  + (size_t)n * H * D + (size_t)hh * D;
  const float* alp = al + (size_t)hh * D;
  const float* arp = ar + (size_t)hh * D;
  float sl = 0.f, sr = 0.f;
  #pragma unroll 4
  for (int d = 0; d < D; ++d) { float v = hp[d]; sl += v * alp[d]; sr += v * arp[d]; }
  el[t] = sl; er[t] = sr;
}

__global__ void gat_fill_state(float* __restrict__ mmax, float* __restrict__ den, int n) {
  int t = blockIdx.x * blockDim.x + threadIdx.x;
  if (t < n) { mmax[t] = -__builtin_inff(); den[t] = 0.f; }
}

// rst[n, j] = b[j]; grid: (ceil(S/256), Nn)
__global__ void gat_init_bias(float* __restrict__ rst, const float* __restrict__ b,
                              int S) {
  int j = blockIdx.x * blockDim.x + threadIdx.x;
  if (j < S) rst[(size_t)blockIdx.y * S + j] = b[j];
}

__global__ void gat_edge_max(const int* __restrict__ src, const int* __restrict__ dst,
                             const float* __restrict__ el, const float* __restrict__ er,
                             float* __restrict__ mmax, int E_, int H) {
  int t = blockIdx.x * blockDim.x + threadIdx.x;
  if (t >= E_ * H) return;
  int e = t / H, hh = t % H;
  int s = src[e], d = dst[e];
  float v = lrelu(el[s * H + hh] + er[d * H + hh]);
  atomicMaxF(&mmax[d * H + hh], v);
}

__global__ void gat_edge_den(const int* __restrict__ src, const int* __restrict__ dst,
                             const float* __restrict__ el, const float* __restrict__ er,
                             const float* __restrict__ mmax, float* __restrict__ den,
                             int E_, int H) {
  int t = blockIdx.x * blockDim.x + threadIdx.x;
  if (t >= E_ * H) return;
  int e = t / H, hh = t % H;
  int s = src[e], d = dst[e];
  float v = lrelu(el[s * H + hh] + er[d * H + hh]);
  atomicAdd(&den[d * H + hh], expf(v - mmax[d * H + hh]));
}

__global__ void gat_edge_aggregate(const int* __restrict__ src, const int* __restrict__ dst,
                                   const float* __restrict__ el, const float* __restrict__ er,
                                   const float* __restrict__ mmax, const float* __restrict__ den,
                                   const float* __restrict__ h, float* __restrict__ rst,
                                   int E_, int H, int D) {
  int t = blockIdx.x * blockDim.x + threadIdx.x;
  if (t >= E_ * H) return;
  int e = t / H, hh = t % H;
  int s = src[e], d = dst[e];
  int ih = d * H + hh;
  float v = lrelu(el[s * H + hh] + er[ih]);
  float alpha = expf(v - mmax[ih]) / den[ih];
  const float* hs = h   + (size_t)s * H * D + (size_t)hh * D;
  float*       rd = rst + (size_t)d * H * D + (size_t)hh * D;
  #pragma unroll 4
  for (int dd = 0; dd < D; ++dd) atomicAdd(&rd[dd], hs[dd] * alpha);
}

// ---------------------------------------------------------------------------
static void run_gat_layer(const float* x, const int* src, const int* dst,
                          const float* W, const float* al, const float* ar,
                          const float* b, float* h, float* el, float* er,
                          float* mm, float* dn, float* rst,
                          int Nn, int K, int H, int D, int E_, hipStream_t stream) {
  const int Nc = H * D;
  dim3 gg((unsigned)((Nn + 127) / 128), (unsigned)((Nc + 63) / 64));
  gat_gemm_f32_wmma<<<gg, 256, 0, stream>>>(x, W, h, Nn, K, Nc);

  int tnh = Nn * H;
  gat_attn_coeff<<<(tnh + 255) / 256, 256, 0, stream>>>(h, al, ar, el, er, Nn, H, D);
  gat_fill_state<<<(tnh + 255) / 256, 256, 0, stream>>>(mm, dn, tnh);
  dim3 gb((unsigned)((Nc + 255) / 256), (unsigned)Nn);
  gat_init_bias<<<gb, 256, 0, stream>>>(rst, b, Nc);

  int te = E_ * H;
  gat_edge_max<<<(te + 255) / 256, 256, 0, stream>>>(src, dst, el, er, mm, E_, H);
  gat_edge_den<<<(te + 255) / 256, 256, 0, stream>>>(src, dst, el, er, mm, dn, E_, H);
  gat_edge_aggregate<<<(te + 255) / 256, 256, 0, stream>>>(src, dst, el, er, mm, dn,
                                                           h, rst, E_, H, D);
}

extern "C" void kernel_launch(void* const* d_in, const int* in_sizes, int n_in,
                              void* d_out, int out_size, void* d_ws, size_t ws_size,
                              hipStream_t stream) {
  const float* x   = (const float*)d_in[0];
  const int*   src = (const int*)  d_in[1];
  const int*   dst = (const int*)  d_in[2];
  const float* W0  = (const float*)d_in[3];
  const float* al0 = (const float*)d_in[4];
  const float* ar0 = (const float*)d_in[5];
  const float* b0  = (const float*)d_in[6];
  const float* W1  = (const float*)d_in[7];
  const float* al1 = (const float*)d_in[8];
  const float* ar1 = (const float*)d_in[9];
  const float* b1  = (const float*)d_in[10];
  const float* W2  = (const float*)d_in[11];
  const float* al2 = (const float*)d_in[12];
  const float* ar2 = (const float*)d_in[13];
  const float* b2  = (const float*)d_in[14];

  const int IN    = 256;
  const int HEADS = 4;
  const int HID   = 64;
  const int Nn = in_sizes[0] / IN;              // 50000
  const int E_ = in_sizes[1];                   // 800000
  const int Cc = in_sizes[11] / (HEADS * HID);  // 47

  float* ws   = (float*)d_ws;
  float* bufA = ws;                                 // [N,256] h (GEMM out)
  float* bufB = bufA + (size_t)Nn * 256;            // [N,256] rst / next input
  float* el   = bufB + (size_t)Nn * 256;            // [N,H]
  float* er   = el   + (size_t)Nn * HEADS;
  float* mm   = er   + (size_t)Nn * HEADS;
  float* dn   = mm   + (size_t)Nn * HEADS;

  // Layer 0: x[N,256] -> h(bufA) -> rst(bufB)
  run_gat_layer(x,    src, dst, W0, al0, ar0, b0, bufA, el, er, mm, dn, bufB,
                Nn, IN, HEADS, HID, E_, stream);
  // Layer 1: bufB -> h(bufA) -> rst(bufB)   (GEMM drains bufB before bias-init)
  run_gat_layer(bufB, src, dst, W1, al1, ar1, b1, bufA, el, er, mm, dn, bufB,
                Nn, HEADS * HID, HEADS, HID, E_, stream);
  // Layer 2: bufB -> h2(bufA, [N,47]) -> rst = d_out (mean over 1 head = id)
  run_gat_layer(bufB, src, dst, W2, al2, ar2, b2, bufA, el, er, mm, dn,
                (float*)d_out, Nn, HEADS * HID, 1, Cc, E_, stream);
}